// TrilinearAttentionLayer_52355651338502
// MI455X (gfx1250) — compile-verified
//
#include <hip/hip_runtime.h>

// TrilinearAttentionLayer for MI455X (gfx1250).
// Bandwidth-bound (128 MiB stream of `response`, ~134 MFLOP): do the length-256
// dot products on the f32 WMMA pipe (V_WMMA_F32_16X16X4_F32) with the per-batch
// vector v_b broadcast across the N dimension. Two-pass deterministic softmax.

typedef float v2f __attribute__((ext_vector_type(2)));
typedef float v8f __attribute__((ext_vector_type(8)));

#define T_DIM 4096
#define D_DIM 256
#define EPS_F 1e-7f

// Kernel 1: out[b,t] = exp(c_b + response[b,t,:] . v_b)
// grid = (T/256, B), block = 256 threads (8 waves x 2 tiles of 16 t's).
__global__ __launch_bounds__(256)
void trilinear_states_kernel(const float* __restrict__ response,
                             const float* __restrict__ prompt,
                             const float* __restrict__ W,
                             float* __restrict__ out) {
  __shared__ float s_v[D_DIM];    // v_b[d] = w_r[d] + prompt[b,d]*w_x[d]
  __shared__ float s_red[256];    // reduction scratch for c_b
  __shared__ float s_st[256];     // per-(wave,tile) 16 states, disjoint regions

  const int tid   = threadIdx.x;
  const int b     = blockIdx.y;
  const int tBase = blockIdx.x * 256;

  // Build v_b and the c_b partial products (one d per thread).
  {
    const float p = prompt[b * D_DIM + tid];
    s_v[tid]   = W[D_DIM + tid] + p * W[2 * D_DIM + tid];
    s_red[tid] = p * W[tid];
  }
  __syncthreads();
  // Fixed-order tree reduction -> c_b (deterministic, no fp atomics).
  #pragma unroll
  for (int s = 128; s > 0; s >>= 1) {
    if (tid < s) s_red[tid] += s_red[tid + s];
    __syncthreads();
  }
  const float c_b = s_red[0];

  const int lane = tid & 31;
  const int wv   = tid >> 5;
  const int half = lane >> 4;   // K half of the wave (ISA 16x4 f32 A layout)
  const int m    = lane & 15;   // A-matrix row (t within tile)

  for (int i = 0; i < 2; ++i) {
    const int tile = wv * 2 + i;
    const int t0   = tBase + tile * 16;
    // A: lane holds response[b, t0+m, 4c + 2*half + {0,1}] per K-chunk c.
    const float* rowp =
        response + ((size_t)b * T_DIM + (size_t)(t0 + m)) * D_DIM + 2 * half;

    v8f acc = {};
    #pragma unroll 8
    for (int c = 0; c < 64; ++c) {
      v2f a  = *(const v2f*)(rowp + 4 * c);                 // 8B global load
      v2f bv = *(const v2f*)(&s_v[4 * c + 2 * half]);       // 8B LDS broadcast
      // D(16x16,f32) = A(16x4,f32) x B(4x16,f32) + C ; B[k][n] = v_b[4c+k] for all n.
      acc = __builtin_amdgcn_wmma_f32_16x16x4_f32(
          /*neg_a=*/false, a, /*neg_b=*/false, bv,
          /*c_mod=*/(short)0, acc, /*reuse_a=*/false, /*reuse_b=*/false);
    }

    // C/D layout: VGPR r, lanes 0-15 -> M=r, lanes 16-31 -> M=8+r (all N equal).
    // Lanes with N==0 (lanes 0 and 16) stage their 8 rows into LDS.
    if (m == 0) {
      #pragma unroll
      for (int r = 0; r < 8; ++r) s_st[tile * 16 + half * 8 + r] = acc[r];
    }
    __syncthreads();  // uniform control flow; orders LDS store -> load
    if (half == 0)    // lanes 0..15: one coalesced 64B store of exp(states)
      out[(size_t)b * T_DIM + (size_t)(t0 + m)] = expf(s_st[tile * 16 + m] + c_b);
  }
}

// Kernel 2: per-batch normalize out[b,:] /= (sum_t out[b,t] + EPS).
// One block per batch; fixed-order partial sums + tree reduction = deterministic.
__global__ __launch_bounds__(256)
void softmax_norm_kernel(float* __restrict__ out) {
  __shared__ float s_red[256];
  const int tid = threadIdx.x;
  float* p = out + (size_t)blockIdx.x * T_DIM;

  float s = 0.f;
  #pragma unroll
  for (int i = 0; i < T_DIM / 256; ++i) s += p[tid + i * 256];
  s_red[tid] = s;
  __syncthreads();
  #pragma unroll
  for (int k = 128; k > 0; k >>= 1) {
    if (tid < k) s_red[tid] += s_red[tid + k];
    __syncthreads();
  }
  const float inv = 1.0f / (s_red[0] + EPS_F);
  #pragma unroll
  for (int i = 0; i < T_DIM / 256; ++i) p[tid + i * 256] *= inv;
}

extern "C" void kernel_launch(void* const* d_in, const int* in_sizes, int n_in,
                              void* d_out, int out_size, void* d_ws, size_t ws_size,
                              hipStream_t stream) {
  const float* response = (const float*)d_in[0];  // (B, T, D) f32
  const float* prompt   = (const float*)d_in[1];  // (B, D)    f32
  const float* W        = (const float*)d_in[2];  // (3D, 1)   f32
  float* out            = (float*)d_out;          // (B, T, 1) f32

  const int B = in_sizes[1] / D_DIM;  // 32

  dim3 grid1(T_DIM / 256, B);
  trilinear_states_kernel<<<grid1, 256, 0, stream>>>(response, prompt, W, out);
  softmax_norm_kernel<<<B, 256, 0, stream>>>(out);
}